// HeadVer3_14035953123648
// MI455X (gfx1250) — compile-verified
//
#include <hip/hip_runtime.h>

typedef __attribute__((ext_vector_type(2))) float v2f;
typedef __attribute__((ext_vector_type(8))) float v8f;

#define B_ 4
#define T_ 4096
#define C_ 1024
#define WAVES 8
#define ROWS_PER_WAVE (T_ / WAVES) /* 512 */

// out[b,t,c] = (1/(t+1)) * sum_{s<=t} x[b,s,c]
// One block = one (b, 16-column strip). 8 waves split T; intra-chunk prefix
// via chained V_WMMA_F32_16X16X4_F32 with L16 (lower-triangular ones) as A.
__global__ __launch_bounds__(256) void causal_uniform_attn(
    const float* __restrict__ x, float* __restrict__ out) {
  __shared__ float wave_sums[WAVES * 16];

  const int tid  = threadIdx.x;
  const int lane = tid & 31;
  const int w    = tid >> 5;       // wave id 0..7
  const int col  = lane & 15;      // column within strip / M row for A
  const int half = lane >> 4;      // lane half selects K/M sub-block

  const int b  = blockIdx.x >> 6;          // C_/16 == 64 strips per batch
  const int c0 = (blockIdx.x & 63) << 4;

  const float* xb = x   + (size_t)b * T_ * C_ + c0 + col;
  float*       ob = out + (size_t)b * T_ * C_ + c0 + col;

  // ---- Phase 1: per-wave column sums over its 512-row chunk ----
  const int tbeg = w * ROWS_PER_WAVE;
  float part = 0.f;
  for (int t = tbeg + half; t < tbeg + ROWS_PER_WAVE; t += 2)
    part += xb[(size_t)t * C_];
  float tot = part + __shfl_xor(part, 16, 32);   // combine lane halves
  if (lane < 16) wave_sums[w * 16 + lane] = tot;
  __syncthreads();

  // ---- Phase 2: exclusive scan of wave totals -> running offset r ----
  float r = 0.f;
  for (int v = 0; v < w; ++v) r += wave_sums[v * 16 + col];

  // ---- A matrices: K-slices of the 16x16 lower-triangular ones ----
  // A 16x4 f32 layout: lanes0-15 {K=0,1}, lanes16-31 {K=2,3}; M = lane%16.
  v2f a[4];
#pragma unroll
  for (int k = 0; k < 4; ++k) {
#pragma unroll
    for (int j = 0; j < 2; ++j) {
      int kk = 4 * k + 2 * half + j;
      a[k][j] = (kk <= col) ? 1.0f : 0.0f;
    }
  }

  // ---- Phase 3: WMMA chunk scan, 16 rows at a time ----
  for (int t0 = tbeg; t0 < tbeg + ROWS_PER_WAVE; t0 += 16) {
    // B 4x16 f32 layout: V0 = {K0 | K2}, V1 = {K1 | K3}; N = lane%16.
    v2f bk[4];
#pragma unroll
    for (int k = 0; k < 4; ++k) {
      int row = t0 + 4 * k + 2 * half;
      bk[k][0] = xb[(size_t)row * C_];
      bk[k][1] = xb[(size_t)(row + 1) * C_];
    }

    v8f d = {r, r, r, r, r, r, r, r};  // broadcast running offset into C
#pragma unroll
    for (int k = 0; k < 4; ++k)
      d = __builtin_amdgcn_wmma_f32_16x16x4_f32(
          /*neg_a=*/false, a[k], /*neg_b=*/false, bk[k],
          /*c_mod=*/(short)0, d, /*reuse_a=*/false, /*reuse_b=*/false);

    // D 16x16: VGPR v = rows {v | v+8}. Scale by 1/(t+1) at store time
    // (single v_rcp_f32 instead of a full IEEE divide sequence).
#pragma unroll
    for (int v = 0; v < 8; ++v) {
      int t = t0 + v + 8 * half;
      ob[(size_t)t * C_] = d[v] * __builtin_amdgcn_rcpf((float)(t + 1));
    }

    // New running sum = row 15 of D (d[7], lanes 16-31), unscaled.
    r = __shfl(d[7], 16 + col, 32);
  }
}

extern "C" void kernel_launch(void* const* d_in, const int* /*in_sizes*/, int /*n_in*/,
                              void* d_out, int /*out_size*/, void* /*d_ws*/, size_t /*ws_size*/,
                              hipStream_t stream) {
  const float* x = (const float*)d_in[0];
  float* out = (float*)d_out;
  dim3 grid(B_ * (C_ / 16));  // 256 blocks
  dim3 block(256);            // 8 waves
  hipLaunchKernelGGL(causal_uniform_attn, grid, block, 0, stream, x, out);
}